// SingelGnn_1073741824404
// MI455X (gfx1250) — compile-verified
//
#include <hip/hip_runtime.h>
#include <hip/hip_bf16.h>

#define NN 40000
#define NE 640000
#define D  128

typedef __attribute__((ext_vector_type(2))) float v2f;
typedef __attribute__((ext_vector_type(4))) float v4f;
typedef __attribute__((ext_vector_type(8))) float v8f;

// ---------------------------------------------------------------------------
// Edge scatter: one wave32 per edge. Each lane handles 4 contiguous floats
// (32 lanes * 4 = 128 = D). edge_attrs is streamed once (non-temporal) so it
// does not evict the L2-resident node features / accumulator. Atomic f32 adds
// resolve in L2 (agg is 20.5 MB, fits easily in the 192 MB L2).
// ---------------------------------------------------------------------------
__global__ __launch_bounds__(256) void gine_scatter(
    const float* __restrict__ x,        // [NN, D] current node features
    const int*   __restrict__ src_idx,  // [NE]
    const int*   __restrict__ dst_idx,  // [NE]
    const float* __restrict__ ea,       // [NE, D] edge attributes
    float*       __restrict__ agg,      // [NN, D] pre-initialized with x
    int n_edges)
{
    int wave = (int)((blockIdx.x * blockDim.x + threadIdx.x) >> 5);
    int lane = threadIdx.x & 31;
    if (wave >= n_edges) return;

    int s = src_idx[wave];
    int d = dst_idx[wave];

    const v4f* xs = (const v4f*)(x  + (size_t)s    * D);
    const v4f* es = (const v4f*)(ea + (size_t)wave * D);

    v4f xv = xs[lane];                                  // L2-hot gather
    v4f ev = __builtin_nontemporal_load(&es[lane]);     // streamed once

    v4f m;
    m.x = fmaxf(xv.x + ev.x, 0.0f);
    m.y = fmaxf(xv.y + ev.y, 0.0f);
    m.z = fmaxf(xv.z + ev.z, 0.0f);
    m.w = fmaxf(xv.w + ev.w, 0.0f);

    float* ag = agg + (size_t)d * D + lane * 4;
    __hip_atomic_fetch_add(ag + 0, m.x, __ATOMIC_RELAXED, __HIP_MEMORY_SCOPE_AGENT);
    __hip_atomic_fetch_add(ag + 1, m.y, __ATOMIC_RELAXED, __HIP_MEMORY_SCOPE_AGENT);
    __hip_atomic_fetch_add(ag + 2, m.z, __ATOMIC_RELAXED, __HIP_MEMORY_SCOPE_AGENT);
    __hip_atomic_fetch_add(ag + 3, m.w, __ATOMIC_RELAXED, __HIP_MEMORY_SCOPE_AGENT);
}

// ---------------------------------------------------------------------------
// out = relu(A @ W + bias), A:[NN,D], W:[D,D], fp32 WMMA 16x16x4.
// One wave per 16x16 output tile; block = 8 waves = all 8 column tiles of one
// 16-row stripe (so the 8 waves share the same A rows through L0/WGP$).
// Grid = NN/16 = 2500 blocks. K loop: 128 / 4 = 32 WMMA ops.
//
// ISA VGPR layouts (CDNA5 ISA):
//   A 16x4 f32 : lane L -> m = L%16, holds K = 2*(L/16) and 2*(L/16)+1
//   B 4x16 f32 : lane L -> n = L%16, VGPR v holds K = v + 2*(L/16)
//   C/D 16x16  : lane L -> n = L%16, VGPR r holds m = r + 8*(L/16)
// ---------------------------------------------------------------------------
__global__ __launch_bounds__(256) void gine_gemm_bias_relu(
    const float* __restrict__ A,     // [NN, D]  (x + aggregated messages)
    const float* __restrict__ W,     // [D, D]   row-major W[k][n]
    const float* __restrict__ bias,  // [D]
    float*       __restrict__ out)   // [NN, D]
{
    const int wavei = threadIdx.x >> 5;   // 0..7 -> column tile
    const int lane  = threadIdx.x & 31;
    const int m0 = blockIdx.x * 16;
    const int n0 = wavei * 16;
    const int g  = lane >> 4;             // lane half (0/1)
    const int r  = lane & 15;

    const float* arow = A + (size_t)(m0 + r) * D;

    v8f c = {};
    #pragma unroll
    for (int k = 0; k < D; k += 4) {
        const int kb = k + 2 * g;
        v2f a = *(const v2f*)(arow + kb);          // A[m][kb], A[m][kb+1]
        v2f b;
        b.x = W[(size_t)kb * D + n0 + r];          // B row kb   (cache-hot, 64 KB)
        b.y = W[(size_t)(kb + 1) * D + n0 + r];    // B row kb+1
        c = __builtin_amdgcn_wmma_f32_16x16x4_f32(
                /*neg_a=*/false, a, /*neg_b=*/false, b,
                /*c_mod=*/(short)0, c, /*reuse_a=*/false, /*reuse_b=*/false);
    }

    const int col = n0 + r;
    const float bv = bias[col];
    #pragma unroll
    for (int i = 0; i < 8; ++i) {
        const int row = m0 + i + 8 * g;
        out[(size_t)row * D + col] = fmaxf(c[i] + bv, 0.0f);
    }
}

// ---------------------------------------------------------------------------
// Driver: two GINE layers. agg is seeded with x via async d2d copy so the
// scatter's atomics produce (x + sum relu(x[src]+ea)) directly.
// ---------------------------------------------------------------------------
extern "C" void kernel_launch(void* const* d_in, const int* in_sizes, int n_in,
                              void* d_out, int out_size, void* d_ws, size_t ws_size,
                              hipStream_t stream) {
    const float* node_feats = (const float*)d_in[0];
    const int*   edge_index = (const int*)  d_in[1];
    const float* edge_attrs = (const float*)d_in[2];
    const float* W1         = (const float*)d_in[3];
    const float* b1         = (const float*)d_in[4];
    const float* W2         = (const float*)d_in[5];
    const float* b2         = (const float*)d_in[6];
    float* out = (float*)d_out;

    float* agg = (float*)d_ws;                       // NN*D floats
    float* h1  = agg + (size_t)NN * D;               // NN*D floats

    const int* src = edge_index;            // row 0 of [2, NE]
    const int* dst = edge_index + NE;       // row 1

    const size_t nd_bytes = (size_t)NN * D * sizeof(float);
    const dim3 sblk(256);
    const dim3 sgrd((NE + 7) / 8);          // 8 edges (waves) per block
    const dim3 gblk(256);
    const dim3 ggrd(NN / 16);               // 2500 row stripes

    // ---- layer 1 ----
    (void)hipMemcpyAsync(agg, node_feats, nd_bytes, hipMemcpyDeviceToDevice, stream);
    gine_scatter<<<sgrd, sblk, 0, stream>>>(node_feats, src, dst, edge_attrs, agg, NE);
    gine_gemm_bias_relu<<<ggrd, gblk, 0, stream>>>(agg, W1, b1, h1);

    // ---- layer 2 ----
    (void)hipMemcpyAsync(agg, h1, nd_bytes, hipMemcpyDeviceToDevice, stream);
    gine_scatter<<<sgrd, sblk, 0, stream>>>(h1, src, dst, edge_attrs, agg, NE);
    gine_gemm_bias_relu<<<ggrd, gblk, 0, stream>>>(agg, W2, b2, out);
}